// PillarEncoder_68659347194293
// MI455X (gfx1250) — compile-verified
//
#include <hip/hip_runtime.h>
#include <hip/hip_bf16.h>
#include <math.h>

// ---------------- problem constants (match reference) ----------------
#define PNUM   48000
#define NPTS   32
#define CIN    4
#define COUT   64
#define XGRID  432
#define YGRID  496
#define BATCH  4
#define VX     0.16f
#define VY     0.16f
#define XOFF   (0.16f * 0.5f + 0.0f)
#define YOFF   (0.16f * 0.5f + (-39.68f))
#define BN_EPS 1e-5f
#define MROWS  (PNUM * NPTS)            // 1,536,000 rows through the MLP
#define INV_M  (1.0f / (float)MROWS)

typedef float v2f __attribute__((ext_vector_type(2)));
typedef float v8f __attribute__((ext_vector_type(8)));

// ---------------------------------------------------------------------
// K0: zero the BN stats accumulators (must happen every launch)
// ---------------------------------------------------------------------
__global__ void k0_zero_stats(float* stats) {
    int t = threadIdx.x;
    if (t < 256) stats[t] = 0.0f;
}

// ---------------------------------------------------------------------
// K1: build 6-ch features + GEMM1 (6 -> 64) + BN1 stat accumulation.
// One wave (32 lanes) == one pillar (32 points). 8 pillars / block.
// ---------------------------------------------------------------------
__global__ __launch_bounds__(256) void k1_feats_gemm1(
    const float* __restrict__ pillars,      // (P, N, 4)
    const int*   __restrict__ coors,        // (P, 4) : b, x, y, z
    const int*   __restrict__ npp,          // (P,)
    const float* __restrict__ w1,           // (64, 6)
    float*       __restrict__ Y,            // (P*N, 64) out: pre-BN1 acts
    float*       __restrict__ stats)        // [0:64] sum1, [64:128] sq1
{
    __shared__ float w1s[COUT * 6];
    __shared__ float ssum[COUT];
    __shared__ float ssq[COUT];

    const int t    = threadIdx.x;
    const int wave = t >> 5;
    const int lane = t & 31;

    for (int i = t; i < COUT * 6; i += 256) w1s[i] = w1[i];
    if (t < COUT) { ssum[t] = 0.0f; ssq[t] = 0.0f; }
    __syncthreads();

    const int p = blockIdx.x * 8 + wave;    // grid = P/8 exactly

    const float* pt = pillars + ((long)p * NPTS + lane) * CIN;
    const float x = pt[0], y = pt[1], z = pt[2];

    // pillar centroid: sum of (x,y,z) over ALL 32 points / npoints
    float sx = x, sy = y, sz = z;
    #pragma unroll
    for (int m = 16; m > 0; m >>= 1) {
        sx += __shfl_xor(sx, m);
        sy += __shfl_xor(sy, m);
        sz += __shfl_xor(sz, m);
    }
    const int   np      = npp[p];
    const float inv_npf = 1.0f / (float)np;
    const float cx = sx * inv_npf, cy = sy * inv_npf, cz = sz * inv_npf;

    const int gx = coors[p * 4 + 1];
    const int gy = coors[p * 4 + 2];

    float f[6];
    f[0] = z;
    f[1] = x - cx;
    f[2] = y - cy;
    f[3] = z - cz;
    f[4] = x - ((float)gx * VX + XOFF);
    f[5] = y - ((float)gy * VY + YOFF);
    const float mask = (lane < np) ? 1.0f : 0.0f;
    #pragma unroll
    for (int c = 0; c < 6; ++c) f[c] *= mask;

    float* yrow = Y + ((long)p * NPTS + lane) * COUT;
    for (int o = 0; o < COUT; ++o) {
        const float* w = &w1s[o * 6];
        float v = f[0]*w[0] + f[1]*w[1] + f[2]*w[2]
                + f[3]*w[3] + f[4]*w[4] + f[5]*w[5];
        yrow[o] = v;
        // wave-level reduction of sum / sumsq for channel o
        float s = v, q = v * v;
        #pragma unroll
        for (int m = 16; m > 0; m >>= 1) {
            s += __shfl_xor(s, m);
            q += __shfl_xor(q, m);
        }
        if (lane == 0) {
            atomicAdd(&ssum[o], s);
            atomicAdd(&ssq[o], q);
        }
    }
    __syncthreads();
    if (t < COUT) {
        atomicAdd(&stats[t],      ssum[t]);
        atomicAdd(&stats[64 + t], ssq[t]);
    }
}

// ---------------------------------------------------------------------
// K2: BN1 + ReLU + GEMM2 (64 -> 64) via V_WMMA_F32_16X16X4_F32,
//     written back in place, accumulating BN2 stats.
// One wave owns a 16-row tile; 8 waves / block; grid = MROWS/16/8.
// ---------------------------------------------------------------------
__global__ __launch_bounds__(256) void k2_bn1_relu_gemm2(
    const float* __restrict__ w2,           // (64, 64) [o][c]
    const float* __restrict__ g1,
    const float* __restrict__ b1,
    float*       __restrict__ Y,            // in: y1, out: y2 (in place)
    float*       __restrict__ stats)        // [128:192] sum2, [192:256] sq2
{
    __shared__ float w2s[COUT * COUT];      // 16 KB
    __shared__ float hA[8][16 * COUT];      // 32 KB: bn1+relu tiles per wave
    __shared__ float sc1[COUT], sh1[COUT];
    __shared__ float ssum[COUT], ssq[COUT];

    const int t    = threadIdx.x;
    const int wave = t >> 5;
    const int lane = t & 31;

    for (int i = t; i < COUT * COUT; i += 256) w2s[i] = w2[i];
    if (t < COUT) {
        const float mean = stats[t] * INV_M;
        const float var  = stats[64 + t] * INV_M - mean * mean;
        const float s    = g1[t] * rsqrtf(var + BN_EPS);
        sc1[t] = s;
        sh1[t] = b1[t] - mean * s;
        ssum[t] = 0.0f;
        ssq[t]  = 0.0f;
    }
    __syncthreads();

    const long tile_m = ((long)blockIdx.x * 8 + wave) * 16;
    float* h = hA[wave];

    // stage bn1(relu) 16x64 tile into LDS (wave-local, LDS pipe is in-order)
    for (int i = lane; i < 16 * COUT; i += 32) {
        const int c = i & 63;
        float v = Y[tile_m * COUT + i];
        v = fmaxf(v * sc1[c] + sh1[c], 0.0f);
        h[i] = v;
    }

    const int half = lane >> 4;             // 0: K+{0,1}, 1: K+{2,3}
    const int l16  = lane & 15;

    for (int nt = 0; nt < 4; ++nt) {
        v8f acc = {};
        #pragma unroll
        for (int kk = 0; kk < 16; ++kk) {
            const int k0  = kk * 4 + half * 2;
            const int col = nt * 16 + l16;
            v2f a, b;
            // A: 16x4 f32 tile (M = l16, K = k0, k0+1)
            a.x = h[l16 * COUT + k0];
            a.y = h[l16 * COUT + k0 + 1];
            // B: 4x16 f32 tile, B[k][n] = w2[n][k]
            b.x = w2s[col * COUT + k0];
            b.y = w2s[col * COUT + k0 + 1];
            acc = __builtin_amdgcn_wmma_f32_16x16x4_f32(
                false, a, false, b, (short)0, acc, false, false);
        }
        // D layout: vgpr r -> row (tile_m + half*8 + r), col nt*16 + l16
        const int gc = nt * 16 + l16;
        float* outp = Y + (tile_m + half * 8) * COUT + gc;
        float s = 0.0f, q = 0.0f;
        #pragma unroll
        for (int r = 0; r < 8; ++r) {
            const float v = acc[r];
            outp[r * COUT] = v;
            s += v;
            q += v * v;
        }
        atomicAdd(&ssum[gc], s);
        atomicAdd(&ssq[gc],  q);
    }
    __syncthreads();
    if (t < COUT) {
        atomicAdd(&stats[128 + t], ssum[t]);
        atomicAdd(&stats[192 + t], ssq[t]);
    }
}

// ---------------------------------------------------------------------
// K3: BN2 + max-pool over the 32 points -> pooled (P, 64)
// ---------------------------------------------------------------------
__global__ __launch_bounds__(256) void k3_bn2_maxpool(
    const float* __restrict__ Y,
    const float* __restrict__ stats,
    const float* __restrict__ g2,
    float*       __restrict__ pooled)
{
    const int idx = blockIdx.x * blockDim.x + threadIdx.x;
    if (idx >= PNUM * COUT) return;
    const int p = idx >> 6;
    const int c = idx & 63;

    const float mean = stats[128 + c] * INV_M;
    const float var  = stats[192 + c] * INV_M - mean * mean;
    const float s    = g2[c] * rsqrtf(var + BN_EPS);
    const float sh   = -mean * s;            // b2 == 0 in reference

    const float* base = Y + ((long)p * NPTS) * COUT + c;
    float m = -INFINITY;
    #pragma unroll 8
    for (int n = 0; n < NPTS; ++n)
        m = fmaxf(m, base[n * COUT] * s + sh);
    pooled[idx] = m;
}

// ---------------------------------------------------------------------
// K4: zero the BEV canvas (float4 grid-stride)
// ---------------------------------------------------------------------
__global__ __launch_bounds__(256) void k4_zero_canvas(float4* __restrict__ out,
                                                      long n4)
{
    const long stride = (long)gridDim.x * blockDim.x;
    for (long i = (long)blockIdx.x * blockDim.x + threadIdx.x; i < n4; i += stride)
        out[i] = make_float4(0.f, 0.f, 0.f, 0.f);
}

// ---------------------------------------------------------------------
// K5: scatter pooled pillar vectors -> out[b][c][y][x]
// ---------------------------------------------------------------------
__global__ __launch_bounds__(256) void k5_scatter(
    const float* __restrict__ pooled,
    const int*   __restrict__ coors,
    float*       __restrict__ out)
{
    const int idx = blockIdx.x * blockDim.x + threadIdx.x;
    if (idx >= PNUM * COUT) return;
    const int p = idx >> 6;
    const int c = idx & 63;
    const int b = coors[p * 4 + 0];
    const int x = coors[p * 4 + 1];
    const int y = coors[p * 4 + 2];
    out[(((long)b * COUT + c) * YGRID + y) * XGRID + x] = pooled[idx];
}

// ---------------------------------------------------------------------
extern "C" void kernel_launch(void* const* d_in, const int* in_sizes, int n_in,
                              void* d_out, int out_size, void* d_ws, size_t ws_size,
                              hipStream_t stream)
{
    const float* pillars = (const float*)d_in[0];
    const int*   coors   = (const int*)  d_in[1];
    const int*   npp     = (const int*)  d_in[2];
    // d_in[3] = batch_size (compile-time BATCH=4)
    const float* w1      = (const float*)d_in[4];
    const float* g1      = (const float*)d_in[5];
    const float* b1      = (const float*)d_in[6];
    const float* w2      = (const float*)d_in[7];
    const float* g2      = (const float*)d_in[8];
    float*       out     = (float*)d_out;

    // workspace layout
    float* stats  = (float*)d_ws;           // 256 floats (BN1/BN2 sum & sumsq)
    float* pooled = stats + 256;            // P*64 floats
    float* Y      = pooled + (long)PNUM * COUT; // P*N*64 floats (y1 -> y2)

    k0_zero_stats<<<1, 256, 0, stream>>>(stats);

    k1_feats_gemm1<<<PNUM / 8, 256, 0, stream>>>(pillars, coors, npp, w1, Y, stats);

    // MROWS / 16 rows-per-wave / 8 waves-per-block = 12000 blocks
    k2_bn1_relu_gemm2<<<MROWS / 128, 256, 0, stream>>>(w2, g1, b1, Y, stats);

    k3_bn2_maxpool<<<(PNUM * COUT + 255) / 256, 256, 0, stream>>>(Y, stats, g2, pooled);

    const long n4 = (long)out_size / 4;     // out_size = 4*64*496*432, /4 exact
    k4_zero_canvas<<<2048, 256, 0, stream>>>((float4*)out, n4);

    k5_scatter<<<(PNUM * COUT + 255) / 256, 256, 0, stream>>>(pooled, coors, out);
}